// FocalModulation1d_49263274885930
// MI455X (gfx1250) — compile-verified
//
#include <hip/hip_runtime.h>
#include <hip/hip_bf16.h>
#include <math.h>

// ---------------- problem constants ----------------
#define Bn   8
#define Cn   256
#define Ln   16384            // 2^14
#define BCL  (Bn * Cn * Ln)   // 33,554,432
#define GR   528              // 256(q) + 256(v) + 16(g padded)

typedef __attribute__((ext_vector_type(2))) float v2f;
typedef __attribute__((ext_vector_type(8))) float v8f;

__device__ __forceinline__ float gelu_exact(float v) {
    return 0.5f * v * (1.0f + erff(v * 0.70710678118654752440f));
}

// ---------------- K0: pack combined projection weights ----------------
__global__ __launch_bounds__(256) void prep_kernel(
    const float* __restrict__ qw, const float* __restrict__ qb,
    const float* __restrict__ vw, const float* __restrict__ vb,
    const float* __restrict__ gw, const float* __restrict__ gb,
    float* __restrict__ wc, float* __restrict__ bc)
{
    int i = blockIdx.x * 256 + threadIdx.x;
    if (i < GR * Cn) {
        int r = i >> 8, c = i & 255;
        float v;
        if      (r < 256) v = qw[r * 256 + c];
        else if (r < 512) v = vw[(r - 256) * 256 + c];
        else if (r < 516) v = gw[(r - 512) * 256 + c];
        else              v = 0.0f;
        wc[i] = v;
    }
    if (i < GR) {
        float v;
        if      (i < 256) v = qb[i];
        else if (i < 512) v = vb[i - 256];
        else if (i < 516) v = gb[i - 512];
        else              v = 0.0f;
        bc[i] = v;
    }
}

// ---------------- K1: fused q/v/g projection GEMM via WMMA f32 16x16x4 ----
// D[e,l] = sum_c Wc[e,c] * x[b,c,l] + bc[e]
// Each wave: 1 M-tile (16 rows) x 4 N-tiles (64 contiguous l) -> A reused 4x.
// A (16x4):  lane m = lane&15, kbase = (lane>>4)*2, regs = {K=kb, K=kb+1}
// B (4x16):  lane n = lane&15, kbase = (lane>>4)*2, regs = {K=kb, K=kb+1}
// D (16x16): reg r -> M = r + (lane>>4)*8, N = lane&15
__global__ __launch_bounds__(256) void proj_gemm_kernel(
    const float* __restrict__ x, const float* __restrict__ wc,
    const float* __restrict__ bc, float* __restrict__ sigq,
    float* __restrict__ focus, float* __restrict__ gates)
{
    const int lane = threadIdx.x & 31;
    const int wave = threadIdx.x >> 5;
    const int e0   = blockIdx.x * 16;                  // 0..512 step 16 (33 tiles)
    const int m    = lane & 15;
    const int kb   = (lane >> 4) << 1;                 // 0 or 2
    const int n16  = lane & 15;
    const int pos  = (blockIdx.y * 8 + wave) * 64;     // 64 l-positions per wave
    const int b    = pos >> 14;                        // / Ln (64 | Ln: no crossing)
    const int l0   = pos & (Ln - 1);
    const float* __restrict__ xb = x + ((size_t)b * Cn) * Ln + l0 + n16;
    const float* __restrict__ wr = wc + (size_t)(e0 + m) * Cn + kb;

    v8f acc[4] = {};
    for (int c0 = 0; c0 < Cn; c0 += 4) {
        v2f a = *(const v2f*)(wr + c0);
        const float* __restrict__ xk0 = xb + (size_t)(c0 + kb) * Ln;
        const float* __restrict__ xk1 = xk0 + Ln;
        #pragma unroll
        for (int j = 0; j < 4; ++j) {
            v2f bm;
            bm.x = xk0[j * 16];
            bm.y = xk1[j * 16];
            acc[j] = __builtin_amdgcn_wmma_f32_16x16x4_f32(
                false, a, false, bm, (short)0, acc[j], false, false);
        }
    }

    const int mhi = (lane >> 4) << 3;                  // 0 or 8
    float bias[8];
    #pragma unroll
    for (int r = 0; r < 8; ++r) bias[r] = bc[e0 + r + mhi];

    // e0 is wave-uniform: pick the destination once (no divergent EXEC churn)
    if (e0 < 256) {                                    // -> sigmoid(query)
        #pragma unroll
        for (int j = 0; j < 4; ++j) {
            #pragma unroll
            for (int r = 0; r < 8; ++r) {
                int e = e0 + r + mhi;
                float v = acc[j][r] + bias[r];
                sigq[((size_t)b * Cn + e) * Ln + l0 + j * 16 + n16] =
                    1.0f / (1.0f + expf(-v));
            }
        }
    } else if (e0 < 512) {                             // -> focus (v proj)
        #pragma unroll
        for (int j = 0; j < 4; ++j) {
            #pragma unroll
            for (int r = 0; r < 8; ++r) {
                int e = e0 + r + mhi - 256;
                focus[((size_t)b * Cn + e) * Ln + l0 + j * 16 + n16] =
                    acc[j][r] + bias[r];
            }
        }
    } else {                                           // -> gates (rows 512..515)
        #pragma unroll
        for (int j = 0; j < 4; ++j) {
            #pragma unroll
            for (int r = 0; r < 8; ++r) {
                int g = r + mhi;                       // 0..15, valid < 4
                if (g < 4) {
                    gates[((size_t)b * 4 + g) * Ln + l0 + j * 16 + n16] =
                        acc[j][r] + bias[r];
                }
            }
        }
    }
}

// ---------------- K2: fused 3-level depthwise conv chain + gating ---------
#define TILE 512
__global__ __launch_bounds__(256) void focal_conv_kernel(
    const float* __restrict__ focus, const float* __restrict__ gates,
    const float* __restrict__ k0w, const float* __restrict__ k1w,
    const float* __restrict__ k2w, float* __restrict__ accb,
    float* __restrict__ meanbuf)
{
    __shared__ float s0[TILE + 12];
    __shared__ float s1[TILE + 10];
    __shared__ float s2[TILE + 6];
    __shared__ float red[256];

    const int t  = threadIdx.x;
    const int c  = blockIdx.y;
    const int b  = blockIdx.z;
    const int l0 = blockIdx.x * TILE;
    const float* __restrict__ fr = focus + ((size_t)b * Cn + c) * Ln;

    for (int i = t; i < TILE + 12; i += 256) {
        int l = l0 - 6 + i;
        s0[i] = (l >= 0 && l < Ln) ? fr[l] : 0.0f;
    }
    float w0[3], w1[5], w2[7];
    #pragma unroll
    for (int j = 0; j < 3; ++j) w0[j] = k0w[c * 3 + j];
    #pragma unroll
    for (int j = 0; j < 5; ++j) w1[j] = k1w[c * 5 + j];
    #pragma unroll
    for (int j = 0; j < 7; ++j) w2[j] = k2w[c * 7 + j];
    __syncthreads();

    // level 0 (k=3): f1 on [l0-5, l0+TILE+5)
    for (int i = t; i < TILE + 10; i += 256) {
        float a = w0[0] * s0[i] + w0[1] * s0[i + 1] + w0[2] * s0[i + 2];
        s1[i] = gelu_exact(a);
    }
    __syncthreads();
    // level 1 (k=5): f2 on [l0-3, l0+TILE+3)
    for (int i = t; i < TILE + 6; i += 256) {
        float a = 0.0f;
        #pragma unroll
        for (int j = 0; j < 5; ++j) a += w1[j] * s1[i + j];
        s2[i] = gelu_exact(a);
    }
    __syncthreads();
    // level 2 (k=7) + gated accumulation + partial mean of f3
    const float* __restrict__ g0 = gates + ((size_t)b * 4 + 0) * Ln;
    const float* __restrict__ g1 = gates + ((size_t)b * 4 + 1) * Ln;
    const float* __restrict__ g2 = gates + ((size_t)b * 4 + 2) * Ln;
    float* __restrict__ ao = accb + ((size_t)b * Cn + c) * Ln;
    float lsum = 0.0f;
    for (int i = t; i < TILE; i += 256) {
        float a = 0.0f;
        #pragma unroll
        for (int j = 0; j < 7; ++j) a += w2[j] * s2[i + j];
        float f3 = gelu_exact(a);
        int l = l0 + i;
        ao[l] = s1[i + 5] * g0[l] + s2[i + 3] * g1[l] + f3 * g2[l];
        lsum += f3;
    }
    red[t] = lsum;
    __syncthreads();
    for (int s = 128; s > 0; s >>= 1) {
        if (t < s) red[t] += red[t + s];
        __syncthreads();
    }
    if (t == 0) atomicAdd(&meanbuf[b * Cn + c], red[0]);
}

// ---------------- K3: global focus = gelu(mean/L) -------------------------
__global__ __launch_bounds__(256) void gf_kernel(float* __restrict__ meanbuf)
{
    int i = blockIdx.x * 256 + threadIdx.x;
    if (i < Bn * Cn) meanbuf[i] = gelu_exact(meanbuf[i] * (1.0f / (float)Ln));
}

// ---------------- K4: mix GEMM via WMMA, gate3 term fused into B-load -----
__global__ __launch_bounds__(256) void mix_gemm_kernel(
    const float* __restrict__ accb, const float* __restrict__ gf,
    const float* __restrict__ gates, const float* __restrict__ mixw,
    const float* __restrict__ mixb, float* __restrict__ macc)
{
    const int lane = threadIdx.x & 31;
    const int wave = threadIdx.x >> 5;
    const int e0   = blockIdx.x * 16;                  // 0..255 (16 tiles)
    const int m    = lane & 15;
    const int kb   = (lane >> 4) << 1;
    const int n16  = lane & 15;
    const int pos  = (blockIdx.y * 8 + wave) * 64;
    const int b    = pos >> 14;
    const int l0   = pos & (Ln - 1);

    const float* __restrict__ ab  = accb + ((size_t)b * Cn) * Ln + l0 + n16;
    const float* __restrict__ g3p = gates + ((size_t)b * 4 + 3) * Ln + l0 + n16;
    const float* __restrict__ gfb = gf + b * Cn;
    const float* __restrict__ wr  = mixw + (size_t)(e0 + m) * Cn + kb;

    float g3v[4];
    #pragma unroll
    for (int j = 0; j < 4; ++j) g3v[j] = g3p[j * 16];

    v8f acc[4] = {};
    for (int c0 = 0; c0 < Cn; c0 += 4) {
        v2f a = *(const v2f*)(wr + c0);
        const int cA = c0 + kb;
        const float gf0 = gfb[cA];
        const float gf1 = gfb[cA + 1];
        const float* __restrict__ a0 = ab + (size_t)cA * Ln;
        const float* __restrict__ a1 = a0 + Ln;
        #pragma unroll
        for (int j = 0; j < 4; ++j) {
            v2f bm;
            bm.x = a0[j * 16] + gf0 * g3v[j];
            bm.y = a1[j * 16] + gf1 * g3v[j];
            acc[j] = __builtin_amdgcn_wmma_f32_16x16x4_f32(
                false, a, false, bm, (short)0, acc[j], false, false);
        }
    }

    const int mhi = (lane >> 4) << 3;
    float bias[8];
    #pragma unroll
    for (int r = 0; r < 8; ++r) bias[r] = mixb[e0 + r + mhi];
    #pragma unroll
    for (int j = 0; j < 4; ++j) {
        #pragma unroll
        for (int r = 0; r < 8; ++r) {
            int e = e0 + r + mhi;
            macc[((size_t)b * Cn + e) * Ln + l0 + j * 16 + n16] =
                acc[j][r] + bias[r];
        }
    }
}

// ---------------- K5: softmax row stats (max, sum of exp) -----------------
__global__ __launch_bounds__(256) void softmax_stats_kernel(
    const float* __restrict__ macc, float* __restrict__ rowmax,
    float* __restrict__ rowsum)
{
    const int row = blockIdx.x;                        // 0..2047 = b*Cn + o
    const int t = threadIdx.x;
    const float* __restrict__ p = macc + (size_t)row * Ln;
    __shared__ float red[256];
    __shared__ float smax;

    float m = -3.4e38f;
    for (int i = t; i < Ln; i += 256) m = fmaxf(m, p[i]);
    red[t] = m;
    __syncthreads();
    for (int s = 128; s > 0; s >>= 1) {
        if (t < s) red[t] = fmaxf(red[t], red[t + s]);
        __syncthreads();
    }
    if (t == 0) smax = red[0];
    __syncthreads();
    float rmax = smax;

    float ssum = 0.0f;
    for (int i = t; i < Ln; i += 256) ssum += expf(p[i] - rmax);
    __syncthreads();
    red[t] = ssum;
    __syncthreads();
    for (int s = 128; s > 0; s >>= 1) {
        if (t < s) red[t] += red[t + s];
        __syncthreads();
    }
    if (t == 0) { rowmax[row] = rmax; rowsum[row] = red[0]; }
}

// ---------------- K6: mask = softmax, out = sigq * mask (float4) ----------
__global__ __launch_bounds__(256) void finalize_kernel(
    const float* __restrict__ macc, const float* __restrict__ rowmax,
    const float* __restrict__ rowsum, float* __restrict__ out,
    float* __restrict__ mask)
{
    size_t i = ((size_t)blockIdx.x * 256 + threadIdx.x) * 4;   // 4 | Ln
    int row = (int)(i >> 14);
    float rm = rowmax[row];
    float rs = 1.0f / rowsum[row];
    float4 mv = *(const float4*)(macc + i);
    float4 ov = *(const float4*)(out + i);   // holds sigmoid(query)
    float4 e;
    e.x = expf(mv.x - rm) * rs;
    e.y = expf(mv.y - rm) * rs;
    e.z = expf(mv.z - rm) * rs;
    e.w = expf(mv.w - rm) * rs;
    *(float4*)(mask + i) = e;
    ov.x *= e.x; ov.y *= e.y; ov.z *= e.z; ov.w *= e.w;
    *(float4*)(out + i) = ov;
}

// ---------------- host launcher ------------------------------------------
extern "C" void kernel_launch(void* const* d_in, const int* in_sizes, int n_in,
                              void* d_out, int out_size, void* d_ws, size_t ws_size,
                              hipStream_t stream) {
    const float* x     = (const float*)d_in[0];
    const float* q_w   = (const float*)d_in[1];
    const float* q_b   = (const float*)d_in[2];
    const float* v_w   = (const float*)d_in[3];
    const float* v_b   = (const float*)d_in[4];
    const float* g_w   = (const float*)d_in[5];
    const float* g_b   = (const float*)d_in[6];
    const float* f0_w  = (const float*)d_in[7];
    const float* f1_w  = (const float*)d_in[8];
    const float* f2_w  = (const float*)d_in[9];
    const float* mix_w = (const float*)d_in[10];
    const float* mix_b = (const float*)d_in[11];

    float* out  = (float*)d_out;          // [B,C,L]
    float* mask = out + (size_t)BCL;      // [B,C,L]; used as acc scratch first

    float* ws = (float*)d_ws;
    size_t o = 0;
    float* wc     = ws + o; o += (size_t)GR * Cn;     // 135168
    float* bc     = ws + o; o += GR;                  // 528
    o = (o + 63) & ~(size_t)63;
    float* focus  = ws + o; o += (size_t)BCL;         // reused as macc later
    float* gatesb = ws + o; o += (size_t)Bn * 4 * Ln; // 524288
    float* meanb  = ws + o; o += Bn * Cn;             // 2048
    float* rowmax = ws + o; o += Bn * Cn;
    float* rowsum = ws + o; o += Bn * Cn;

    // K0: pack combined projection weights/bias
    prep_kernel<<<(GR * Cn + 255) / 256, 256, 0, stream>>>(
        q_w, q_b, v_w, v_b, g_w, g_b, wc, bc);

    // zero mean accumulator (graph-capturable memset node)
    hipMemsetAsync(meanb, 0, (size_t)Bn * Cn * sizeof(float), stream);

    // K1: fused q/v/g projections. 33 M-tiles x 2048 wave-groups (4 N-tiles/wave).
    proj_gemm_kernel<<<dim3(33, 256), 256, 0, stream>>>(
        x, wc, bc, out /*sigq*/, focus, gatesb);

    // K2: fused focal conv chain; acc written into mask-half of d_out.
    focal_conv_kernel<<<dim3(Ln / TILE, Cn, Bn), 256, 0, stream>>>(
        focus, gatesb, f0_w, f1_w, f2_w, mask /*acc*/, meanb);

    // K3: global focus
    gf_kernel<<<(Bn * Cn + 255) / 256, 256, 0, stream>>>(meanb);

    // K4: mix GEMM (gate3 fused); macc reuses focus buffer (now dead).
    mix_gemm_kernel<<<dim3(16, 256), 256, 0, stream>>>(
        mask /*acc*/, meanb, gatesb, mix_w, mix_b, focus /*macc*/);

    // K5: softmax stats per (b,channel) row
    softmax_stats_kernel<<<Bn * Cn, 256, 0, stream>>>(focus, rowmax, rowsum);

    // K6: finalize mask and out (float4 vectorized)
    finalize_kernel<<<(BCL / 4 + 255) / 256, 256, 0, stream>>>(
        focus, rowmax, rowsum, out, mask);
}